// FFSlotAttentionEncoder_11639361372393
// MI455X (gfx1250) — compile-verified
//
#include <hip/hip_runtime.h>
#include <hip/hip_bf16.h>

// ---------------------------------------------------------------------------
// FFSlotAttentionEncoder for MI455X (gfx1250, wave32, WMMA).
// One workgroup (256 threads = 8 wave32) per batch row b.
// H[b] (1024x64 f32, 256KB) is materialized entirely in the WGP's 320KB LDS.
// Both MLP GEMMs use V_WMMA_F32_16X16X4_F32.
// W1/W2 are stored in LDS pre-packed in WMMA B-fragment order so each lane
// loads its 2-float B fragment with a single aligned ds_load_b64.
// Each wave processes TWO 16-row S-tiles per iteration so every B fragment
// feeds two WMMAs (halves LDS B traffic, doubles independent WMMA work).
// ---------------------------------------------------------------------------

typedef __attribute__((ext_vector_type(2))) float v2f;
typedef __attribute__((ext_vector_type(8))) float v8f;

static constexpr int Bn   = 1024;   // batch
static constexpr int Sn   = 1024;   // slots
static constexpr int DIN  = 128;
static constexpr int DSL  = 64;
static constexpr int TPB  = 256;    // 8 waves of 32
static constexpr int NT   = 4;      // 4 N-tiles of 16 cover DSL=64
static constexpr int KTOP = 8;

// ---- LDS layout (float slots) ----
static constexpr int H_OFF    = 0;                      // 1024*64
static constexpr int W1_OFF   = H_OFF  + Sn * DSL;      // 128*64 (packed)
static constexpr int W2_OFF   = W1_OFF + DIN * DSL;     // 64*64  (packed)
static constexpr int B1_OFF   = W2_OFF + DSL * DSL;     // 64
static constexpr int B2_OFF   = B1_OFF + 64;            // 64
static constexpr int QV_OFF   = B2_OFF + 64;            // 64
static constexpr int SC_OFF   = QV_OFF + 64;            // 1024 masked scores
static constexpr int AT_OFF   = SC_OFF + Sn;            // 1024 attn weights
static constexpr int RF_OFF   = AT_OFF + Sn;            // 256 reduce (float)
static constexpr int RI_OFF   = RF_OFF + TPB;           // 256 reduce (int)
static constexpr int FLG_OFF  = RI_OFF + TPB;           // any-mask flag
static constexpr int SMEM_FLOATS = FLG_OFF + 4;         // 80580 -> 322320 B < 320KB

extern "C" __global__ __launch_bounds__(TPB, 1)
void ffslot_attn_encoder_kernel(const float* __restrict__ X,   // [B,S,DIN]
                                const float* __restrict__ M,   // [B,S]
                                const float* __restrict__ W1,  // [DIN,DSL]
                                const float* __restrict__ b1,  // [DSL]
                                const float* __restrict__ W2,  // [DSL,DSL]
                                const float* __restrict__ b2,  // [DSL]
                                const float* __restrict__ q,   // [2,DSL]
                                float* __restrict__ out_sel,   // [B,8,DSL]
                                float* __restrict__ out_ctx,   // [B,DSL]
                                float* __restrict__ out_attn)  // [B,S]
{
    extern __shared__ float sm[];
    float* Hs  = sm + H_OFF;
    float* w1s = sm + W1_OFF;   // packed: [(k>>1)*64 + n]*2 + (k&1)
    float* w2s = sm + W2_OFF;   // packed: same transform
    float* b1s = sm + B1_OFF;
    float* b2s = sm + B2_OFF;
    float* qvs = sm + QV_OFF;
    float* scM = sm + SC_OFF;
    float* at  = sm + AT_OFF;
    float* rf  = sm + RF_OFF;
    int*   ri  = (int*)(sm + RI_OFF);
    int*   flg = (int*)(sm + FLG_OFF);

    const int tid  = threadIdx.x;
    const int b    = blockIdx.x;
    const int lane = tid & 31;      // wave32 lane
    const int wave = tid >> 5;      // 0..7
    const int lo   = lane & 15;
    const int hi   = lane >> 4;     // 0/1: ISA A/B layout half-wave select
    const int koff = hi * 2;        // low lanes K=0,1 ; high lanes K=2,3

    // ------------- Preload weights (B-fragment packed), biases, query ------
    for (int i = tid; i < DIN * DSL; i += TPB) {
        const int k = i >> 6, n = i & 63;
        w1s[(((k >> 1) << 6) + n) * 2 + (k & 1)] = W1[i];
    }
    for (int i = tid; i < DSL * DSL; i += TPB) {
        const int k = i >> 6, n = i & 63;
        w2s[(((k >> 1) << 6) + n) * 2 + (k & 1)] = W2[i];
    }
    if (tid < DSL) {
        b1s[tid] = b1[tid];
        b2s[tid] = b2[tid];
        // mean over 2 heads * scale 1/sqrt(64) = *(0.5 * 0.125)
        qvs[tid] = (q[tid] + q[DSL + tid]) * 0.0625f;
    }
    if (tid == 0) flg[0] = 0;
    __syncthreads();

    // ---------------- Phase 1: H = relu(X@W1+b1)@W2+b2 via fp32 WMMA -------
    // Each wave handles tiles {2w, 2w+1} + 16p : 4 dual-tile passes.
    const size_t xbase = (size_t)b * Sn * DIN;
    for (int t0 = wave * 2; t0 < Sn / 16; t0 += 16) {   // uniform per wave
        const int rowA0 = t0 * 16 + lo;
        const float* Ap0 = X + xbase + (size_t)rowA0 * DIN + koff;
        const float* Ap1 = Ap0 + 16 * DIN;
        if (t0 + 16 < Sn / 16)   // prefetch next pass (global_prefetch)
            __builtin_prefetch(X + xbase + (size_t)(rowA0 + 256) * DIN, 0, 0);

        // ---- GEMM1: two 16x128 tiles @ 128x64, K-steps of 4 ----
        v8f acc[2][NT] = {};
#pragma unroll
        for (int kk = 0; kk < DIN / 4; ++kk) {
            v2f a0 = *(const v2f*)(Ap0 + 4 * kk);       // A: 16x4 f32 layout
            v2f a1 = *(const v2f*)(Ap1 + 4 * kk);
            const float* bp = w1s + (((kk * 2 + hi) << 6) << 1);
#pragma unroll
            for (int nt = 0; nt < NT; ++nt) {
                const int n = nt * 16 + lo;
                v2f bv = *(const v2f*)(bp + 2 * n);     // one ds_load_b64
                acc[0][nt] = __builtin_amdgcn_wmma_f32_16x16x4_f32(
                    false, a0, false, bv, (short)0, acc[0][nt], false, false);
                acc[1][nt] = __builtin_amdgcn_wmma_f32_16x16x4_f32(
                    false, a1, false, bv, (short)0, acc[1][nt], false, false);
            }
        }
        // bias + ReLU; park Hmid tiles in their final LDS slots (C layout)
#pragma unroll
        for (int u = 0; u < 2; ++u) {
#pragma unroll
            for (int nt = 0; nt < NT; ++nt) {
                const int n = nt * 16 + lo;
                const float bb = b1s[n];
#pragma unroll
                for (int j = 0; j < 8; ++j)
                    Hs[((t0 + u) * 16 + j + hi * 8) * DSL + n] =
                        fmaxf(acc[u][nt][j] + bb, 0.0f);
            }
        }

        // ---- GEMM2: two Hmid(16x64) tiles @ W2(64x64) ----
        v8f acc2[2][NT] = {};
        const float* Hrow0 = Hs + (t0 * 16 + lo) * DSL + koff;  // same-wave LDS RAW
        const float* Hrow1 = Hrow0 + 16 * DSL;
#pragma unroll
        for (int kk = 0; kk < DSL / 4; ++kk) {
            v2f a0 = *(const v2f*)(Hrow0 + 4 * kk);             // ds_load_b64
            v2f a1 = *(const v2f*)(Hrow1 + 4 * kk);
            const float* bp = w2s + (((kk * 2 + hi) << 6) << 1);
#pragma unroll
            for (int nt = 0; nt < NT; ++nt) {
                const int n = nt * 16 + lo;
                v2f bv = *(const v2f*)(bp + 2 * n);
                acc2[0][nt] = __builtin_amdgcn_wmma_f32_16x16x4_f32(
                    false, a0, false, bv, (short)0, acc2[0][nt], false, false);
                acc2[1][nt] = __builtin_amdgcn_wmma_f32_16x16x4_f32(
                    false, a1, false, bv, (short)0, acc2[1][nt], false, false);
            }
        }
#pragma unroll
        for (int u = 0; u < 2; ++u) {
#pragma unroll
            for (int nt = 0; nt < NT; ++nt) {
                const int n = nt * 16 + lo;
                const float bb = b2s[n];
#pragma unroll
                for (int j = 0; j < 8; ++j)
                    Hs[((t0 + u) * 16 + j + hi * 8) * DSL + n] = acc2[u][nt][j] + bb;
            }
        }
    }
    __syncthreads();

    // ---------------- Phase 2: scores + mask -------------------------------
    const float* mrow = M + (size_t)b * Sn;
    int localAny = 0;
#pragma unroll
    for (int j2 = 0; j2 < Sn / TPB; ++j2) {
        const int s = tid + j2 * TPB;
        const float* hr = Hs + s * DSL;
        float sc = 0.0f;
        for (int n = 0; n < DSL; ++n) sc += hr[n] * qvs[n];
        const bool mk = mrow[s] > 0.5f;
        localAny |= (int)mk;
        scM[s] = mk ? sc : -__builtin_inff();
    }
    if (localAny) atomicOr(flg, 1);
    __syncthreads();
    const int anyM = flg[0];

    // ---------------- Phase 3: masked softmax ------------------------------
    float lmax = -__builtin_inff();
#pragma unroll
    for (int j2 = 0; j2 < Sn / TPB; ++j2) {
        const int s = tid + j2 * TPB;
        lmax = fmaxf(lmax, anyM ? scM[s] : 0.0f);
    }
    rf[tid] = lmax; __syncthreads();
    for (int off = TPB / 2; off > 0; off >>= 1) {
        if (tid < off) rf[tid] = fmaxf(rf[tid], rf[tid + off]);
        __syncthreads();
    }
    const float gmax = rf[0]; __syncthreads();

    float lsum = 0.0f;
#pragma unroll
    for (int j2 = 0; j2 < Sn / TPB; ++j2) {
        const int s = tid + j2 * TPB;
        const float v = anyM ? scM[s] : 0.0f;
        const float e = expf(v - gmax);   // exp(-inf)=0 for masked-out slots
        at[s] = e;
        lsum += e;
    }
    rf[tid] = lsum; __syncthreads();
    for (int off = TPB / 2; off > 0; off >>= 1) {
        if (tid < off) rf[tid] += rf[tid + off];
        __syncthreads();
    }
    const float inv = 1.0f / rf[0]; __syncthreads();

    float* attnRow = out_attn + (size_t)b * Sn;
#pragma unroll
    for (int j2 = 0; j2 < Sn / TPB; ++j2) {
        const int s = tid + j2 * TPB;
        const float w = at[s] * inv;
        at[s] = w;
        attnRow[s] = w;
    }
    __syncthreads();

    // ---------------- Phase 4: ctx = attnW @ H -----------------------------
    {
        const int n = tid & 63;
        const int c = tid >> 6;          // 4 chunks of 256 slots
        float p = 0.0f;
        for (int s = c * 256; s < (c + 1) * 256; ++s) p += at[s] * Hs[s * DSL + n];
        rf[tid] = p; __syncthreads();
        if (tid < DSL)
            out_ctx[(size_t)b * DSL + tid] =
                rf[tid] + rf[tid + 64] + rf[tid + 128] + rf[tid + 192];
        __syncthreads();
    }

    // ---------------- Phase 5: top-8 (stable, lowest-index ties) -----------
    if (!anyM) {   // invalid row: top-k over zeros -> indices 0..7
#pragma unroll
        for (int j2 = 0; j2 < Sn / TPB; ++j2) scM[tid + j2 * TPB] = 0.0f;
    }
    __syncthreads();

    for (int r = 0; r < KTOP; ++r) {
        float bv = -__builtin_inff();
        int   bi = 0x7fffffff;
#pragma unroll
        for (int j2 = 0; j2 < Sn / TPB; ++j2) {
            const int s = tid + j2 * TPB;
            const float v = scM[s];
            if (v > bv || (v == bv && s < bi)) { bv = v; bi = s; }
        }
        rf[tid] = bv; ri[tid] = bi; __syncthreads();
        for (int off = TPB / 2; off > 0; off >>= 1) {
            if (tid < off) {
                const float v2 = rf[tid + off]; const int i2 = ri[tid + off];
                if (v2 > rf[tid] || (v2 == rf[tid] && i2 < ri[tid])) {
                    rf[tid] = v2; ri[tid] = i2;
                }
            }
            __syncthreads();
        }
        const int widx = ri[0];
        if (tid < DSL)
            out_sel[((size_t)b * KTOP + r) * DSL + tid] = Hs[widx * DSL + tid];
        if (tid == 0) scM[widx] = -__builtin_inff();
        __syncthreads();
    }
}

// ---------------------------------------------------------------------------
extern "C" void kernel_launch(void* const* d_in, const int* in_sizes, int n_in,
                              void* d_out, int out_size, void* d_ws, size_t ws_size,
                              hipStream_t stream) {
    (void)in_sizes; (void)n_in; (void)out_size; (void)d_ws; (void)ws_size;
    const float* X  = (const float*)d_in[0];   // slot_feats [B,S,DIN]
    const float* M  = (const float*)d_in[1];   // slot_mask  [B,S]
    const float* W1 = (const float*)d_in[2];   // [DIN,DSL]
    const float* b1 = (const float*)d_in[3];   // [DSL]
    const float* W2 = (const float*)d_in[4];   // [DSL,DSL]
    const float* b2 = (const float*)d_in[5];   // [DSL]
    const float* q  = (const float*)d_in[6];   // [2,DSL]
    // d_in[7] = k (==8, compile-time KTOP)

    float* out      = (float*)d_out;
    float* out_sel  = out;                                         // B*8*64
    float* out_ctx  = out + (size_t)Bn * KTOP * DSL;               // B*64
    float* out_attn = out + (size_t)Bn * KTOP * DSL + (size_t)Bn * DSL; // B*S

    const size_t lds_bytes = (size_t)SMEM_FLOATS * sizeof(float);  // ~315 KB < 320 KB/WGP
    ffslot_attn_encoder_kernel<<<dim3(Bn), dim3(TPB), lds_bytes, stream>>>(
        X, M, W1, b1, W2, b2, q, out_sel, out_ctx, out_attn);
}